// ROIPooling_60258391163280
// MI455X (gfx1250) — compile-verified
//
#include <hip/hip_runtime.h>
#include <math.h>

// Vector types matching the probe-reported tensor_load_to_lds signatures.
typedef unsigned int v4u __attribute__((ext_vector_type(4)));
typedef int          v4i __attribute__((ext_vector_type(4)));
typedef int          v8i __attribute__((ext_vector_type(8)));

#define FH 200          // feature height
#define FW 304          // feature width
#define OH 7
#define OW 7
#define MAX_CROP 180    // proposals in [0,180) -> crop dims <= 180
#define BLOCK 256       // 8 wave32s per workgroup

#if defined(__HIP_DEVICE_COMPILE__) && \
    __has_builtin(__builtin_amdgcn_tensor_load_to_lds) && \
    __has_builtin(__builtin_amdgcn_s_wait_tensorcnt)
#define USE_TDM 1
#else
#define USE_TDM 0
#endif

extern "C" __global__ void __launch_bounds__(BLOCK)
roi_pool_tdm_kernel(const float* __restrict__ feat,
                    const int*   __restrict__ prop,
                    float*       __restrict__ out) {
    // Dynamic LDS layout:
    //   tile[0 .. MAX_CROP*MAX_CROP)          : HxW crop tile (row stride W)
    //   rbm [MAX_CROP*MAX_CROP .. +OH*MAX_CROP): per-(row-bin, column) maxes
    extern __shared__ float tile[];
    float* rbm = tile + MAX_CROP * MAX_CROP;

    const int plane = blockIdx.x;     // (b*256 + c), 0..2047
    const int tid   = threadIdx.x;

    // Proposals are uniform across the grid; force into SGPRs.
    const int x1 = __builtin_amdgcn_readfirstlane(prop[0]);
    const int y1 = __builtin_amdgcn_readfirstlane(prop[1]);
    const int x2 = __builtin_amdgcn_readfirstlane(prop[2]);
    const int y2 = __builtin_amdgcn_readfirstlane(prop[3]);
    const int H  = y2 - y1 + 1;       // 1..180
    const int W  = x2 - x1 + 1;       // 1..180

#if USE_TDM
    // Wave 0 issues one TDM DMA of the whole HxW crop tile into LDS.
    if (tid < 32) {
        const unsigned lds_off = (unsigned)__builtin_amdgcn_groupstaticsize();
        const unsigned long long ga =
            (unsigned long long)(const void*)feat +
            4ull * ((unsigned long long)plane * (unsigned long long)(FH * FW) +
                    (unsigned long long)(y1 * FW + x1));

        // D# group 0 (128b): count=1 | lds_addr | global_addr[56:0] | type=2
        v4u g0 = { 1u,
                   lds_off,
                   (unsigned)ga,
                   (unsigned)((ga >> 32) & 0x1FFFFFFull) | 0x80000000u };

        // D# group 1 (256b):
        //  d0: workgroup_mask=0, data_size=2 (4B), no barrier/iterate/pad
        //  d1[31:16] = tensor_dim0[15:0] = W
        //  d2[31:16] = tensor_dim1[15:0] = H   (tensor_dim0[31:16] = 0)
        //  d3[31:16] = tile_dim0 = W           (tensor_dim1[31:16] = 0)
        //  d4[15:0]  = tile_dim1 = H ; tile_dim2 = 0
        //  d5        = tensor_dim0_stride = 304 (elements)
        v8i g1 = { (int)0x20000u,
                   (int)(((unsigned)W & 0xFFFFu) << 16),
                   (int)(((unsigned)H & 0xFFFFu) << 16),
                   (int)(((unsigned)W & 0xFFFFu) << 16),
                   (int)((unsigned)H & 0xFFFFu),
                   FW,
                   0, 0 };

        v4i zz = { 0, 0, 0, 0 };      // groups 2/3: tile_dim2=0 -> unused
#if defined(__clang_major__) && (__clang_major__ >= 23)
        v8i z8 = { 0, 0, 0, 0, 0, 0, 0, 0 };
        __builtin_amdgcn_tensor_load_to_lds(g0, g1, zz, zz, z8, 0);
#else
        __builtin_amdgcn_tensor_load_to_lds(g0, g1, zz, zz, 0);
#endif
        __builtin_amdgcn_s_wait_tensorcnt(0);   // TENSORcnt -> 0: tile is in LDS
    }
    __syncthreads();
#else
    // Fallback: cooperative copy (also what the host-side parse sees).
    for (int idx = tid; idx < H * W; idx += BLOCK) {
        const int r = idx / W, c = idx % W;
        tile[idx] = feat[(size_t)plane * (FH * FW) + (size_t)(y1 + r) * FW + (x1 + c)];
    }
    __syncthreads();
#endif

    // Phase 1: per-(row-bin, column) maxes. 7*W tasks striped over 256 lanes;
    // consecutive lanes read consecutive columns -> conflict-free LDS access.
    // Row bin i covers [floor(i*H/7), ceil((i+1)*H/7)) (PyTorch adaptive).
    for (int t = tid; t < OH * W; t += BLOCK) {
        const int i = t / W;
        const int c = t - i * W;
        const int r0 = (i * H) / OH;
        const int r1 = ((i + 1) * H + OH - 1) / OH;
        float m = -INFINITY;
        for (int r = r0; r < r1; ++r) m = fmaxf(m, tile[r * W + c]);
        rbm[i * W + c] = m;
    }
    __syncthreads();

    // Phase 2: 49 output bins, one per lane; coalesced store.
    if (tid < OH * OW) {
        const int i  = tid / OW, j = tid % OW;
        const int c0 = (j * W) / OW;
        const int c1 = ((j + 1) * W + OW - 1) / OW;
        const float* row = &rbm[i * W];
        float m = -INFINITY;
        for (int c = c0; c < c1; ++c) m = fmaxf(m, row[c]);
        out[(size_t)plane * (OH * OW) + tid] = m;
    }
}

extern "C" void kernel_launch(void* const* d_in, const int* in_sizes, int n_in,
                              void* d_out, int out_size, void* d_ws, size_t ws_size,
                              hipStream_t stream) {
    const float* feat = (const float*)d_in[0];   // (8,256,200,304) f32
    const int*   prop = (const int*)d_in[1];     // (4,) i32 sorted: x1,y1,x2,y2
    float*       out  = (float*)d_out;           // (8,256,7,7) f32

    const int planes = in_sizes[0] / (FH * FW);  // 2048
    const size_t shmem =
        (size_t)(MAX_CROP * MAX_CROP + OH * MAX_CROP) * sizeof(float); // 134640 B

    hipLaunchKernelGGL(roi_pool_tdm_kernel,
                       dim3(planes), dim3(BLOCK), shmem, stream,
                       feat, prop, out);
}